// VariationalGCNEncoder_91130616086749
// MI455X (gfx1250) — compile-verified
//
#include <hip/hip_runtime.h>

// ---------------------------------------------------------------------------
// VariationalGCNEncoder: N=100000 nodes, E=3.2M edges, 512 -> 4 -> (2,2)
// Memory-bound (~500MB traffic). fp32 WMMA (16x16x4) for the x@W1 GEMM.
// ---------------------------------------------------------------------------

typedef __attribute__((ext_vector_type(2))) float v2f;
typedef __attribute__((ext_vector_type(8))) float v8f;

#define IN_CH 512

// ---- init: deg = 1 (self loop), zero both aggregation buffers -------------
__global__ void k_init(float* __restrict__ deg, float4* __restrict__ agg1,
                       float4* __restrict__ agg2, int n) {
    int i = blockIdx.x * blockDim.x + threadIdx.x;
    if (i < n) {
        deg[i]  = 1.0f;
        agg1[i] = make_float4(0.f, 0.f, 0.f, 0.f);
        agg2[i] = make_float4(0.f, 0.f, 0.f, 0.f);
    }
}

// ---- in-degree count via float atomics ------------------------------------
__global__ void k_degree(const int* __restrict__ dst, float* __restrict__ deg, int e) {
    int i = blockIdx.x * blockDim.x + threadIdx.x;
    if (i < e) atomicAdd(&deg[dst[i]], 1.0f);
}

// ---- dinv = rsqrt(deg)  (deg >= 1 always, self loops included) ------------
__global__ void k_rsqrt(float* __restrict__ deg, int n) {
    int i = blockIdx.x * blockDim.x + threadIdx.x;
    if (i < n) deg[i] = rsqrtf(deg[i]);
}

// ---- xW = x @ W1 via V_WMMA_F32_16X16X4_F32 -------------------------------
// One wave computes a 16-node x 16-col tile (cols 4..15 are zero padding).
// A layout (f32 16x4): lanes 0-15 hold (M=lane, K=k,k+1); lanes 16-31 hold
// (M=lane-16, K=k+2,k+3) -> per-lane b64 load of the node's row.
// B (4x16) is staged in LDS zero-padded to 512x16 so reads are unpredicated.
__global__ void __launch_bounds__(256) k_gemm(const float* __restrict__ x,
                                              const float* __restrict__ W1,
                                              float* __restrict__ xW, int n) {
    __shared__ float sB[IN_CH * 16];   // 32 KB: W1 padded 512x16
    for (int idx = threadIdx.x; idx < IN_CH * 16; idx += blockDim.x) {
        int r = idx >> 4, c = idx & 15;
        sB[idx] = (c < 4) ? W1[r * 4 + c] : 0.0f;
    }
    __syncthreads();

    const int lane = threadIdx.x & 31;
    const int wave = (blockIdx.x * blockDim.x + threadIdx.x) >> 5;
    const int nodeBase = wave * 16;
    if (nodeBase >= n) return;                      // wave-uniform exit

    const int row = lane & 15;                      // A row / B-D column
    const int hi  = lane >> 4;                      // lane-half selects K pair
    int nr = nodeBase + row; if (nr > n - 1) nr = n - 1;
    const float* xrow = x + (size_t)nr * IN_CH + hi * 2;

    v8f acc = {};
    #pragma unroll 8
    for (int k = 0; k < IN_CH; k += 4) {
        v2f a = *(const v2f*)(xrow + k);            // K = k+2*hi, k+2*hi+1
        int br = (k + hi * 2) * 16 + row;
        v2f b;
        b.x = sB[br];                               // B row k+2*hi,   col=row
        b.y = sB[br + 16];                          // B row k+2*hi+1, col=row
        acc = __builtin_amdgcn_wmma_f32_16x16x4_f32(
            false, a, false, b, (short)0, acc, false, false);
    }

    // D layout: lane c (and 16+c) holds column c; VGPR v holds M = v (+8).
    // Full tiles (always the case when 16 | n) take the unpredicated path:
    // hoisting the bounds check to a wave-uniform scalar branch avoids the
    // per-element v_cmpx/exec-juggling tail.
    if (row < 4) {
        float* outp = xW + (size_t)(nodeBase + hi * 8) * 4 + row;
        if (nodeBase + 16 <= n) {                   // wave-uniform fast path
            #pragma unroll
            for (int v = 0; v < 8; ++v) outp[v * 4] = acc[v];
        } else {                                    // ragged last tile
            #pragma unroll
            for (int v = 0; v < 8; ++v) {
                if (nodeBase + hi * 8 + v < n) outp[v * 4] = acc[v];
            }
        }
    }
}

// ---- edge scatter: agg[dst] += h[src] * dinv[src]*dinv[dst] ---------------
__global__ void k_scatter(const int* __restrict__ src, const int* __restrict__ dst,
                          const float* __restrict__ dinv,
                          const float4* __restrict__ hin,
                          float* __restrict__ agg, int e) {
    int i = blockIdx.x * blockDim.x + threadIdx.x;
    if (i >= e) return;
    int s = src[i], d = dst[i];
    float w = dinv[s] * dinv[d];
    float4 m = hin[s];
    atomicAdd(&agg[d * 4 + 0], m.x * w);
    atomicAdd(&agg[d * 4 + 1], m.y * w);
    atomicAdd(&agg[d * 4 + 2], m.z * w);
    atomicAdd(&agg[d * 4 + 3], m.w * w);
}

// ---- h = relu(agg1 + dinv^2 * xW (self loop) + b1) ------------------------
__global__ void k_relu(const float4* __restrict__ xW, const float4* __restrict__ agg1,
                       const float* __restrict__ dinv, const float* __restrict__ b1,
                       float4* __restrict__ h, int n) {
    int i = blockIdx.x * blockDim.x + threadIdx.x;
    if (i >= n) return;
    float wsf = dinv[i] * dinv[i];
    float4 a = agg1[i], xw = xW[i], r;
    r.x = fmaxf(0.f, fmaf(wsf, xw.x, a.x) + b1[0]);
    r.y = fmaxf(0.f, fmaf(wsf, xw.y, a.y) + b1[1]);
    r.z = fmaxf(0.f, fmaf(wsf, xw.z, a.z) + b1[2]);
    r.w = fmaxf(0.f, fmaf(wsf, xw.w, a.w) + b1[3]);
    h[i] = r;
}

// ---- finalize: g = agg2 + dinv^2*h ; mu = g@Wmu+bmu ; ls = g@Wls+bls ------
// (linear layers commute with segment_sum, so one scatter serves both heads)
__global__ void k_out(const float4* __restrict__ h, const float4* __restrict__ agg2,
                      const float* __restrict__ dinv,
                      const float* __restrict__ Wmu, const float* __restrict__ bmu,
                      const float* __restrict__ Wls, const float* __restrict__ bls,
                      float* __restrict__ out, int n) {
    int i = blockIdx.x * blockDim.x + threadIdx.x;
    if (i >= n) return;
    float wsf = dinv[i] * dinv[i];
    float4 a = agg2[i], hh = h[i];
    float g0 = fmaf(wsf, hh.x, a.x);
    float g1 = fmaf(wsf, hh.y, a.y);
    float g2 = fmaf(wsf, hh.z, a.z);
    float g3 = fmaf(wsf, hh.w, a.w);
    float mu0 = g0 * Wmu[0] + g1 * Wmu[2] + g2 * Wmu[4] + g3 * Wmu[6] + bmu[0];
    float mu1 = g0 * Wmu[1] + g1 * Wmu[3] + g2 * Wmu[5] + g3 * Wmu[7] + bmu[1];
    float ls0 = g0 * Wls[0] + g1 * Wls[2] + g2 * Wls[4] + g3 * Wls[6] + bls[0];
    float ls1 = g0 * Wls[1] + g1 * Wls[3] + g2 * Wls[5] + g3 * Wls[7] + bls[1];
    out[(size_t)i * 2 + 0] = mu0;
    out[(size_t)i * 2 + 1] = mu1;
    out[(size_t)2 * n + i * 2 + 0] = ls0;
    out[(size_t)2 * n + i * 2 + 1] = ls1;
}

extern "C" void kernel_launch(void* const* d_in, const int* in_sizes, int n_in,
                              void* d_out, int out_size, void* d_ws, size_t ws_size,
                              hipStream_t stream) {
    const float* x   = (const float*)d_in[0];
    const int*   ei  = (const int*)  d_in[1];   // [2, E] int32
    const float* W1  = (const float*)d_in[2];   // [512, 4]
    const float* b1  = (const float*)d_in[3];   // [4]
    const float* Wmu = (const float*)d_in[4];   // [4, 2]
    const float* bmu = (const float*)d_in[5];   // [2]
    const float* Wls = (const float*)d_in[6];   // [4, 2]
    const float* bls = (const float*)d_in[7];   // [2]
    float* out = (float*)d_out;

    const int n = in_sizes[0] / IN_CH;          // 100000
    const int e = in_sizes[1] / 2;              // 3200000
    const int* srcIdx = ei;                     // edge_index[0]
    const int* dstIdx = ei + e;                 // edge_index[1]

    // workspace layout (16B-aligned slices): xW | h | agg1 | agg2 | dinv
    float4* xW   = (float4*)d_ws;               // n float4
    float4* h    = xW + n;                      // n float4
    float4* agg1 = h + n;                       // n float4
    float4* agg2 = agg1 + n;                    // n float4
    float*  dinv = (float*)(agg2 + n);          // n float

    const int tb = 256;
    const int gn = (n + tb - 1) / tb;
    const int ge = (e + tb - 1) / tb;

    k_init  <<<gn, tb, 0, stream>>>(dinv, agg1, agg2, n);
    k_degree<<<ge, tb, 0, stream>>>(dstIdx, dinv, e);
    k_rsqrt <<<gn, tb, 0, stream>>>(dinv, n);

    const int tiles = (n + 15) / 16;            // one wave per 16-node tile
    const int wavesPerBlock = tb / 32;
    k_gemm<<<(tiles + wavesPerBlock - 1) / wavesPerBlock, tb, 0, stream>>>(
        x, W1, (float*)xW, n);

    k_scatter<<<ge, tb, 0, stream>>>(srcIdx, dstIdx, dinv, xW, (float*)agg1, e);
    k_relu   <<<gn, tb, 0, stream>>>(xW, agg1, dinv, b1, h, n);
    k_scatter<<<ge, tb, 0, stream>>>(srcIdx, dstIdx, dinv, h, (float*)agg2, e);
    k_out    <<<gn, tb, 0, stream>>>(h, agg2, dinv, Wmu, bmu, Wls, bls, out, n);
}